// OverlapLoss_intra_63110249447561
// MI455X (gfx1250) — compile-verified
//
#include <hip/hip_runtime.h>
#include <hip/hip_bf16.h>

#define IMG_W 1440.0f
#define IMG_H 2560.0f

typedef __attribute__((ext_vector_type(2))) float v2f;
typedef __attribute__((ext_vector_type(8))) float v8f;

// One workgroup per batch (B=64). 128 threads = one per odd slot.
// Computes the per-batch sum of IoU(min-area) over all unordered pairs of
// unique-id last-occurrence representatives that share a parent.
__global__ void overlap_intra_batch(const float* __restrict__ pred_boxes,
                                    const int*   __restrict__ idv_g,
                                    const int*   __restrict__ par_g,
                                    float*       __restrict__ partial) {
    __shared__ int    last[128];    // last occurrence index per id value (atomicMax)
    __shared__ int    par[128];     // parent per entry
    __shared__ float4 bx[128];      // xyxy per entry
    __shared__ int    repPar[128];  // parent of representative, -1 if id absent
    __shared__ float4 repBox[128];  // xyxy of representative
    __shared__ float  red[128];

    const int b = blockIdx.x;
    const int t = threadIdx.x;          // entry index m in [0,128)
    const int s = 2 * t + 1;            // odd slot within the batch row
    const int g = b * 256 + s;          // flat [B*S] index

    __builtin_prefetch(&((const float4*)pred_boxes)[g], 0, 0); // global_prefetch_b8

    const int   myId  = idv_g[g];       // in [0,128)
    const int   myPar = par_g[g];       // in [0,16)
    const float4 box  = ((const float4*)pred_boxes)[g];

    // xcycwh -> xyxy exactly like the reference (scale, then +/- half wh)
    const float cx = box.x * IMG_W, cy = box.y * IMG_H;
    const float ww = box.z * IMG_W, hh = box.w * IMG_H;
    float4 xy;
    xy.x = cx - ww * 0.5f;
    xy.y = cy - hh * 0.5f;
    xy.z = cx + ww * 0.5f;
    xy.w = cy + hh * 0.5f;

    last[t] = -1;
    par[t]  = myPar;
    bx[t]   = xy;
    __syncthreads();

    // last occurrence = max entry index holding this id value
    atomicMax(&last[myId], t);
    __syncthreads();

    // Thread t owns id value v = t: gather its representative.
    const int m = last[t];
    if (m >= 0) {
        repPar[t] = par[m];
        repBox[t] = bx[m];
    } else {
        repPar[t] = -1;
        repBox[t] = make_float4(0.f, 0.f, 0.f, 0.f);
    }
    __syncthreads();

    // All unordered pairs (v < u) of present ids with equal parent.
    float acc = 0.0f;
    const int    pr = repPar[t];
    const float4 b1 = repBox[t];
    if (pr >= 0) {
        for (int u = t + 1; u < 128; ++u) {
            if (repPar[u] == pr) {
                const float4 b2 = repBox[u];
                const float xl = fmaxf(b1.x, b2.x);
                const float yt = fmaxf(b1.y, b2.y);
                const float xr = fminf(b1.z, b2.z);
                const float yb = fminf(b1.w, b2.w);
                if (xr >= xl && yb >= yt) {
                    const float inter = (xr - xl) * (yb - yt);
                    const float a1 = (b1.z - b1.x) * (b1.w - b1.y);
                    const float a2 = (b2.z - b2.x) * (b2.w - b2.y);
                    acc += inter / fminf(a1, a2);
                }
            }
        }
    }

    // Deterministic tree reduction within the workgroup.
    red[t] = acc;
    __syncthreads();
    for (int off = 64; off > 0; off >>= 1) {
        if (t < off) red[t] += red[t + off];
        __syncthreads();
    }
    if (t == 0) partial[b] = red[0];
}

// Final reduction of 64 partials with V_WMMA_F32_16X16X4_F32.
// A holds the 64 partials (bijectively packed into the 16x4 operand),
// B is all-ones, C = 0. Every column n of D satisfies
//   sum_m D[m,n] = sum of all A entries = grand total,
// independent of the exact A lane striping. Each lane sums its 8 D VGPRs
// (rows of its half) and adds the opposite half via shfl_xor(16) (wave32).
__global__ void reduce64_wmma(const float* __restrict__ partial,
                              float*       __restrict__ out) {
    const int lane = threadIdx.x;      // 0..31, one full wave, EXEC all ones
    const int m  = lane & 15;
    const int hi = lane >> 4;          // 0: K={0,1}, 1: K={2,3} per ISA A layout

    v2f a;
    a.x = partial[m * 4 + (hi ? 2 : 0)];
    a.y = partial[m * 4 + (hi ? 3 : 1)];
    v2f bb;
    bb.x = 1.0f;
    bb.y = 1.0f;
    v8f c = {};

    // 8 args: (neg_a, A, neg_b, B, c_mod, C, reuse_a, reuse_b)
    v8f d = __builtin_amdgcn_wmma_f32_16x16x4_f32(
        /*neg_a=*/false, a, /*neg_b=*/false, bb,
        /*c_mod=*/(short)0, c, /*reuse_a=*/false, /*reuse_b=*/false);

    float s = d[0] + d[1] + d[2] + d[3] + d[4] + d[5] + d[6] + d[7];
    s += __shfl_xor(s, 16, 32);        // combine M=0..7 half with M=8..15 half
    if (lane == 0) out[0] = s;
}

extern "C" void kernel_launch(void* const* d_in, const int* in_sizes, int n_in,
                              void* d_out, int out_size, void* d_ws, size_t ws_size,
                              hipStream_t stream) {
    const float* pred_boxes = (const float*)d_in[0]; // [64,256,4] f32
    const int*   idp        = (const int*)d_in[1];   // [64,256] i32
    const int*   parp       = (const int*)d_in[2];   // [64,256] i32
    // d_in[3] (type_id) is unused by the reference.

    float* out     = (float*)d_out;
    float* partial = (float*)d_ws;                   // 64 floats of scratch

    const int B = in_sizes[1] / 256;                 // 64

    overlap_intra_batch<<<B, 128, 0, stream>>>(pred_boxes, idp, parp, partial);
    reduce64_wmma<<<1, 32, 0, stream>>>(partial, out);
}